// SpikingCell_87213605912591
// MI455X (gfx1250) — compile-verified
//
#include <hip/hip_runtime.h>
#include <stdint.h>

// SpikingCell scan, MI455X (gfx1250).
// out[b,t,n] depends only on x[b,0..t-1,n] -> independent per-element scalar
// recurrence over T. HBM-bound: ~268MB traffic => ~11.5us floor @ 23.3TB/s.
// CDNA5 path used: global_load_async_to_lds_b64 (ASYNCcnt) as a 32-deep
// software prefetch pipeline along T, drained with s_wait_asynccnt.

#define TSTEPS 256
#define NFEAT  8192
#define BATCH  16

#define VEC    2                    // floats per thread (b64 async / ds_load_b64)
#define BLOCK  128                  // threads per block (4 waves)
#define CHUNK  (BLOCK * VEC)        // 256 features per block
#define NBUF   32                   // prefetch ring depth (asynccnt <= 31)
#define STAGE_BYTES (CHUNK * 4)     // 1KB per stage

typedef float v2f __attribute__((ext_vector_type(2)));

__global__ __launch_bounds__(BLOCK)
void spiking_scan(const float* __restrict__ x, float* __restrict__ out) {
    const int tid   = threadIdx.x;
    const int b     = blockIdx.y;
    const int nbase = blockIdx.x * CHUNK;

    // per-block element base at t=0 of x/out ([B,T,N] row-major)
    const size_t   blkBase = (size_t)b * TSTEPS * NFEAT + (size_t)nbase;
    const uint64_t xblk    = (uint64_t)(uintptr_t)(x + blkBase);
    float* __restrict__ oblk = out + blkBase;

    __shared__ float buf[NBUF][CHUNK];   // 32 KB stage ring

    const unsigned laneByte = (unsigned)(tid * VEC * 4);
    // LDS byte address of this thread's slot in stage 0 (addrspace(3) offset
    // == low 32 bits of the generic pointer)
    const unsigned ldsSlot0 = (unsigned)(uintptr_t)(&buf[0][tid * VEC]);

    // ---- t = 0: dv buffer starts at zero -> spikes are all zero ----
    float v0 = 0.f, v1 = 0.f;
    float r0 = 1.f, r1 = 1.f;            // refrac counters after step 0
    {
        v2f z = {0.f, 0.f};
        __builtin_nontemporal_store(z, (v2f*)(oblk + tid * VEC));
    }

    // ---- prologue: async-prefetch stages 0..NBUF-2 (stage s holds x row s) ----
#pragma unroll
    for (int s = 0; s < NBUF - 1; ++s) {
        unsigned voff = (unsigned)(s * NFEAT * 4) + laneByte;
        unsigned lds  = ldsSlot0 + (unsigned)(s * STAGE_BYTES);
        asm volatile("global_load_async_to_lds_b64 %0, %1, %2"
                     :: "v"(lds), "v"(voff), "s"(xblk) : "memory");
    }

    // ---- steady state: at step t consume x row (t-1), prefetch row t+NBUF-2 ----
    for (int t = 1; t < TSTEPS; ++t) {
        const int s   = t + (NBUF - 2);                    // stage to issue
        const int row = (s < TSTEPS) ? s : (TSTEPS - 1);   // tail: clamp (unused data)
        unsigned voff = (unsigned)(row * NFEAT * 4) + laneByte;
        unsigned lds  = ldsSlot0 + (unsigned)((s & (NBUF - 1)) * STAGE_BYTES);
        asm volatile("global_load_async_to_lds_b64 %0, %1, %2"
                     :: "v"(lds), "v"(voff), "s"(xblk) : "memory");

        // keep NBUF-1 loads in flight; guarantees stage (t-1) has landed in LDS
        asm volatile("s_wait_asynccnt 0x1f" ::: "memory");

        const v2f xv = *reinterpret_cast<const v2f*>(
            &buf[(t - 1) & (NBUF - 1)][tid * VEC]);

        // element 0  (exact mirror of reference ops, fp32)
        float g0 = (r0 >= 2.0f) ? 1.0f : 0.0f;
        v0 += xv.x * g0;
        float o0 = (v0 >= 1.0f) ? 1.0f : 0.0f;
        v0 -= o0;
        r0 = (o0 == 0.0f) ? (r0 + 1.0f) : 0.0f;
        // element 1
        float g1 = (r1 >= 2.0f) ? 1.0f : 0.0f;
        v1 += xv.y * g1;
        float o1 = (v1 >= 1.0f) ? 1.0f : 0.0f;
        v1 -= o1;
        r1 = (o1 == 0.0f) ? (r1 + 1.0f) : 0.0f;

        v2f o = {o0, o1};
        __builtin_nontemporal_store(o, (v2f*)(oblk + (size_t)t * NFEAT + tid * VEC));
    }
}

extern "C" void kernel_launch(void* const* d_in, const int* in_sizes, int n_in,
                              void* d_out, int out_size, void* d_ws, size_t ws_size,
                              hipStream_t stream) {
    (void)in_sizes; (void)n_in; (void)d_ws; (void)ws_size; (void)out_size;
    const float* x = (const float*)d_in[0];
    float* out     = (float*)d_out;

    dim3 grid(NFEAT / CHUNK, BATCH);   // 32 x 16 = 512 blocks, 4 waves each
    spiking_scan<<<grid, BLOCK, 0, stream>>>(x, out);
}